// CustomEmbeddingRegularizer_79121887527439
// MI455X (gfx1250) — compile-verified
//
#include <hip/hip_runtime.h>

#define RATE 0.04f

typedef float v2f __attribute__((ext_vector_type(2)));
typedef float v8f __attribute__((ext_vector_type(8)));

// ---------------------------------------------------------------------------
// lower_bound over sorted int32 edge_src (E <= 1.6M fits int32 indexing)
// ---------------------------------------------------------------------------
__device__ __forceinline__ int lower_bound_i32(const int* __restrict__ a, int n, int key) {
    int lo = 0, hi = n;
    while (lo < hi) {
        int mid = (lo + hi) >> 1;
        if (a[mid] < key) lo = mid + 1; else hi = mid;
    }
    return lo;
}

__global__ void init_out_kernel(float* __restrict__ out) {
    if (blockIdx.x == 0 && threadIdx.x == 0) out[0] = 0.0f;
}

// ---------------------------------------------------------------------------
// Sum of squares: out += RATE * sum(x^2).  Streaming b128 loads, block reduce.
// ---------------------------------------------------------------------------
__global__ void ss_kernel(const float* __restrict__ x, long long total4, long long total,
                          float* __restrict__ out) {
    __shared__ float red[256];
    float acc = 0.0f;
    long long idx    = (long long)blockIdx.x * blockDim.x + threadIdx.x;
    long long stride = (long long)gridDim.x * blockDim.x;
    for (long long i = idx; i < total4; i += stride) {
        float4 v = ((const float4*)x)[i];
        acc = fmaf(v.x, v.x, acc);
        acc = fmaf(v.y, v.y, acc);
        acc = fmaf(v.z, v.z, acc);
        acc = fmaf(v.w, v.w, acc);
    }
    if (idx == 0) {  // scalar tail (<4 elems)
        for (long long i = total4 * 4; i < total; ++i) acc = fmaf(x[i], x[i], acc);
    }
    red[threadIdx.x] = acc;
    __syncthreads();
    for (int s = 128; s > 0; s >>= 1) {
        if ((int)threadIdx.x < s) red[threadIdx.x] += red[threadIdx.x + s];
        __syncthreads();
    }
    if (threadIdx.x == 0) atomicAdd(out, RATE * red[0]);
}

// ---------------------------------------------------------------------------
// Edge kernel: one wave32 per tile of 16 edges.
//   C(16x16) = A(16x128) @ B(16x128)^T via 32x V_WMMA_F32_16X16X4_F32,
//   keep diagonal, scale by 1/deg(src) (deg via binary search on sorted src),
//   out += -RATE * sum.
// D is fixed at 128 floats per row.
// ---------------------------------------------------------------------------
__global__ void edge_kernel(const float* __restrict__ X,
                            const int* __restrict__ esrc,
                            const int* __restrict__ edst,
                            int E, float* __restrict__ out) {
    const int lane = threadIdx.x & 31;
    const int wid  = threadIdx.x >> 5;
    const int tile = blockIdx.x * (blockDim.x >> 5) + wid;
    const int ntiles = (E + 15) >> 4;
    __shared__ float wred[8];

    float contrib = 0.0f;
    if (tile < ntiles) {                  // wave-uniform: EXEC all-ones inside
        const int n = lane & 15;          // edge slot within tile (A row m == B col n)
        int e = tile * 16 + n;
        const bool valid = (e < E);
        if (!valid) e = E - 1;            // clamp: contributes with scale 0

        const int s  = esrc[e];
        const int dn = edst[e];

        // 1/deg(src) — only lanes 0..15 search, then broadcast to upper half
        float scale = 0.0f;
        if (lane < 16) {
            const int lo  = lower_bound_i32(esrc, E, s);
            const int hi  = lower_bound_i32(esrc, E, s + 1);
            const int deg = hi - lo;      // >= 1 (edge e itself)
            scale = valid ? (1.0f / (float)deg) : 0.0f;
        }
        scale = __shfl(scale, n, 32);

        const float* __restrict__ xs = X + (size_t)s  * 128;
        const float* __restrict__ xd = X + (size_t)dn * 128;
        const int khalf = (lane >= 16) ? 4 : 0;   // upper lanes supply upper K half

        v8f c = {};
        #pragma unroll
        for (int u = 0; u < 16; ++u) {
            const int off = 8 * u + khalf;
            float4 av = *(const float4*)(xs + off);   // A: src rows (b128/lane)
            float4 bv = *(const float4*)(xd + off);   // B: dst rows (b128/lane)
            v2f a0 = {av.x, av.y};
            v2f b0 = {bv.x, bv.y};
            c = __builtin_amdgcn_wmma_f32_16x16x4_f32(false, a0, false, b0,
                                                      (short)0, c, false, false);
            v2f a1 = {av.z, av.w};
            v2f b1 = {bv.z, bv.w};
            c = __builtin_amdgcn_wmma_f32_16x16x4_f32(false, a1, false, b1,
                                                      (short)0, c, false, false);
        }

        // Diagonal of C: (m,m) sits at lane m / VGPR m (m<8) and
        // lane m+16 / VGPR m-8 (m>=8); lane's own scale (n = lane&15) matches.
        float diag = 0.0f;
        #pragma unroll
        for (int r = 0; r < 8; ++r) {
            const float cr = c[r];
            diag += ((lane == r) || (lane == r + 24)) ? cr : 0.0f;
        }
        contrib = diag * scale;
    }

    // wave32 reduce
    #pragma unroll
    for (int off = 16; off > 0; off >>= 1)
        contrib += __shfl_xor(contrib, off, 32);
    if (lane == 0) wred[wid] = contrib;
    __syncthreads();

    if (threadIdx.x == 0) {
        float bsum = 0.0f;
        const int nw = (int)(blockDim.x >> 5);
        for (int i = 0; i < nw; ++i) bsum += wred[i];
        atomicAdd(out, -RATE * bsum);
    }
}

// ---------------------------------------------------------------------------
extern "C" void kernel_launch(void* const* d_in, const int* in_sizes, int n_in,
                              void* d_out, int out_size, void* d_ws, size_t ws_size,
                              hipStream_t stream) {
    (void)n_in; (void)out_size; (void)d_ws; (void)ws_size;

    const float* X    = (const float*)d_in[0];
    const int*   esrc = (const int*)d_in[1];
    const int*   edst = (const int*)d_in[2];

    const long long total = (long long)in_sizes[0];   // N * 128
    const int       E     = in_sizes[1];

    float* out = (float*)d_out;

    init_out_kernel<<<1, 1, 0, stream>>>(out);

    const long long total4 = total / 4;
    long long wantBlocks = (total4 + 255) / 256;
    int ssBlocks = (wantBlocks > 4096) ? 4096 : (wantBlocks < 1 ? 1 : (int)wantBlocks);
    ss_kernel<<<ssBlocks, 256, 0, stream>>>(X, total4, total, out);

    const int ntiles = (E + 15) / 16;
    const int wavesPerBlock = 8;                       // 256 threads
    int blocks = (ntiles + wavesPerBlock - 1) / wavesPerBlock;
    if (blocks < 1) blocks = 1;
    edge_kernel<<<blocks, 256, 0, stream>>>(X, esrc, edst, E, out);
}